// DehazeFilter_32478542692835
// MI455X (gfx1250) — compile-verified
//
#include <hip/hip_runtime.h>

typedef __attribute__((ext_vector_type(2))) float v2f;
typedef __attribute__((ext_vector_type(8))) float v8f;

#define HH 2048
#define WW 2048
#define HWTOT (HH * WW)
#define RDARK 7
#define RGUIDE 60
#define EPSG 1e-4f
#define OMEGA_C 0.95f

__device__ __forceinline__ int imin(int a, int b) { return a < b ? a : b; }
__device__ __forceinline__ int imax(int a, int b) { return a > b ? a : b; }

// ---------------- utility ----------------
__global__ void k_zero(unsigned* p, int n) {
  int i = blockIdx.x * blockDim.x + threadIdx.x;
  if (i < n) p[i] = 0u;
}

// ---------------- dark channel ----------------
__global__ void k_cmin(const float* __restrict__ x, float* __restrict__ out) {
  int i = blockIdx.x * blockDim.x + threadIdx.x;  // float4 index
  const float4* a = (const float4*)x;
  const float4* b = (const float4*)(x + HWTOT);
  const float4* c = (const float4*)(x + 2 * HWTOT);
  float4 va = a[i], vb = b[i], vc = c[i], r;
  r.x = fminf(va.x, fminf(vb.x, vc.x));
  r.y = fminf(va.y, fminf(vb.y, vc.y));
  r.z = fminf(va.z, fminf(vb.z, vc.z));
  r.w = fminf(va.w, fminf(vb.w, vc.w));
  ((float4*)out)[i] = r;
}

__global__ void k_hmin(const float* __restrict__ in, float* __restrict__ out) {
  int i = blockIdx.x * blockDim.x + threadIdx.x;
  if (i >= HWTOT) return;
  int y = i >> 11, x = i & (WW - 1);
  float m = 1e30f;
#pragma unroll
  for (int d = -RDARK; d <= RDARK; ++d) {
    int xx = imax(0, imin(WW - 1, x + d));  // clamped dup == +inf pad for min
    m = fminf(m, in[y * WW + xx]);
  }
  out[i] = m;
}

__global__ void k_vmin(const float* __restrict__ in, float* __restrict__ out) {
  int i = blockIdx.x * blockDim.x + threadIdx.x;
  if (i >= HWTOT) return;
  int y = i >> 11, x = i & (WW - 1);
  float m = 1e30f;
#pragma unroll
  for (int d = -RDARK; d <= RDARK; ++d) {
    int yy = imax(0, imin(HH - 1, y + d));
    m = fminf(m, in[yy * WW + x]);
  }
  out[i] = m;
}

// ---------------- atmospheric light (histogram select) ----------------
__global__ void k_hist(const float* __restrict__ dc, unsigned* __restrict__ hist) {
  int i0 = (blockIdx.x * blockDim.x + threadIdx.x) * 8;
#pragma unroll
  for (int j = 0; j < 8; ++j) {
    float v = dc[i0 + j];
    int b = (int)(v * 65536.0f);
    b = b < 0 ? 0 : (b > 65535 ? 65535 : b);
    atomicAdd(&hist[b], 1u);
  }
}

__global__ void k_thresh(const unsigned* __restrict__ hist, float* __restrict__ wsf) {
  __shared__ unsigned part[256];
  int t = threadIdx.x;
  unsigned s = 0;
  for (int i = 0; i < 256; ++i) s += hist[t * 256 + i];
  part[t] = s;
  __syncthreads();
  if (t == 0) {
    const unsigned need = (unsigned)(HWTOT / 1000) + 1u;  // (idx+1)-th largest
    unsigned cum = 0;
    int chunk = 255;
    for (; chunk > 0; --chunk) {
      if (cum + part[chunk] >= need) break;
      cum += part[chunk];
    }
    int bin = chunk * 256 + 255;
    for (; bin > chunk * 256; --bin) {
      cum += hist[bin];
      if (cum >= need) break;
    }
    wsf[0] = (float)bin * (1.0f / 65536.0f);  // lower edge of bin holding s[idx]
  }
}

__global__ void k_masked(const float* __restrict__ x, const float* __restrict__ dc,
                         const float* __restrict__ wsf, float* __restrict__ acc) {
  float thres = wsf[0];
  int i0 = (blockIdx.x * blockDim.x + threadIdx.x) * 16;
  float c = 0.f, s0 = 0.f, s1 = 0.f, s2 = 0.f;
#pragma unroll
  for (int j = 0; j < 16; ++j) {
    int i = i0 + j;
    if (dc[i] >= thres) {
      c += 1.f;
      s0 += x[i];
      s1 += x[HWTOT + i];
      s2 += x[2 * HWTOT + i];
    }
  }
  if (c > 0.f) {
    atomicAdd(&acc[0], c);
    atomicAdd(&acc[1], s0);
    atomicAdd(&acc[2], s1);
    atomicAdd(&acc[3], s2);
  }
}

__global__ void k_avg(float* wsf) {
  if (threadIdx.x == 0 && blockIdx.x == 0) {
    float cnt = wsf[2];
    float a0 = wsf[3] / cnt, a1 = wsf[4] / cnt, a2 = wsf[5] / cnt;
    wsf[1] = 0.299f * a0 + 0.587f * a1 + 0.114f * a2;
  }
}

// ---------------- horizontal box sums (r=60) with WMMA-based workgroup scan ----
// One block per row (256 thr = 8 waves). For each of 10 quantities: per-thread
// serial prefix of 8 elems -> 256 partial totals -> inclusive scan of the 16x16
// totals matrix via D = L*X using 4 chained V_WMMA_F32_16X16X4_F32 (L = lower-
// triangular ones). All waves run the WMMA (EXEC must be all-1s); wave 0
// extracts D + cross-group offsets. Box sum = P[min(x+60,W-1)] - P[x-61].
__global__ void k_hbox(const float* __restrict__ x, const float* __restrict__ dc,
                       const float* __restrict__ wsf, float* __restrict__ planes) {
  __shared__ alignas(16) float sIn[4][WW];
  __shared__ float sP[WW];
  __shared__ float sTot[256];
  __shared__ float sScan[256];
  __shared__ float sG[16];
  const int t = threadIdx.x;
  const int y = blockIdx.x;
  // global->LDS staging as b128 (rows are 8KB-aligned), conflict-free LDS writes
#pragma unroll
  for (int j = 0; j < 2; ++j) {
    int v4 = t + j * 256;  // float4 index in 512-float4 row
    ((float4*)sIn[0])[v4] = ((const float4*)(x + 0 * HWTOT + y * WW))[v4];
    ((float4*)sIn[1])[v4] = ((const float4*)(x + 1 * HWTOT + y * WW))[v4];
    ((float4*)sIn[2])[v4] = ((const float4*)(x + 2 * HWTOT + y * WW))[v4];
    ((float4*)sIn[3])[v4] = ((const float4*)(dc + y * WW))[v4];
  }
  const float kom = OMEGA_C / wsf[1];  // t = 1 - kom * dc
  __syncthreads();

  const int lane = t & 31;
  const int half = (lane >> 4) & 1;
  const int n = lane & 15;  // A: row M; B/D: column N (group index)
  const int base = t * 8;

  for (int q = 0; q < 10; ++q) {
    float pr[8];
    float s = 0.f;
#pragma unroll
    for (int j = 0; j < 8; ++j) {
      int xx = base + j;
      float v;
      if (q < 3)
        v = sIn[q][xx];
      else if (q == 3)
        v = 1.f - kom * sIn[3][xx];
      else if (q < 7)
        v = (1.f - kom * sIn[3][xx]) * sIn[q - 4][xx];
      else {
        float u = sIn[q - 7][xx];
        v = u * u;
      }
      s += v;
      pr[j] = s;
    }
    sTot[t] = s;
    __syncthreads();

    // D(16x16) = L(16x16) x X(16x16), X[k][g] = sTot[16g+k], via 4 K=4 WMMAs
    v8f d;
#pragma unroll
    for (int v = 0; v < 8; ++v) d[v] = 0.f;
#pragma unroll
    for (int c = 0; c < 4; ++c) {
      const int k0 = 4 * c + (half ? 2 : 0);  // per documented 16x4 f32 A layout
      const int k1 = 4 * c + (half ? 3 : 1);
      v2f a, b;
      a.x = (k0 <= n) ? 1.f : 0.f;  // L[M][k] = (k <= M)
      a.y = (k1 <= n) ? 1.f : 0.f;
      b.x = sTot[n * 16 + k0];      // B[k][N] = X[4c+kk][N]
      b.y = sTot[n * 16 + k1];
      d = __builtin_amdgcn_wmma_f32_16x16x4_f32(false, a, false, b, (short)0, d,
                                                false, false);
    }
    if (t < 32) {                       // wave 0 extracts; others were EXEC filler
      if (half) sG[n] = d[7];           // group sums: M=15 row (VGPR7, hi lanes)
      float off = 0.f;
      for (int g = 0; g < n; ++g) off += sG[g];  // exclusive cross-group offset
#pragma unroll
      for (int v = 0; v < 8; ++v) sScan[n * 16 + v + (half ? 8 : 0)] = d[v] + off;
    }
    __syncthreads();

    const float excl = (t == 0) ? 0.f : sScan[t - 1];
#pragma unroll
    for (int j = 0; j < 8; ++j) sP[base + j] = pr[j] + excl;
    __syncthreads();

    // strided b32 stores: lane-consecutive addresses coalesce to 128B/wave and
    // keep the shifted sP reads bank-conflict-free
    float* outq = planes + q * HWTOT + y * WW;
#pragma unroll
    for (int j = 0; j < 8; ++j) {
      int xx = t + j * 256;
      float sb = sP[imin(xx + RGUIDE, WW - 1)];
      if (xx > RGUIDE) sb -= sP[xx - RGUIDE - 1];
      outq[xx] = sb;
    }
    __syncthreads();
  }
}

// ---------------- vertical box sums fused with guided-filter math + dehaze ----
// Block = 128 threads x 2 adjacent columns (float2 traffic); grid (8, 16).
// Running 121-row window per plane; leaving-row reloads hit L2 (~10 MB set).
__device__ __forceinline__ void dehaze_px(const float S[10], float invN, float avg,
                                          float i0, float i1, float i2, float& o0,
                                          float& o1, float& o2) {
  const float mp = S[3] * invN;
  float iv[3] = {i0, i1, i2};
  float r[3];
#pragma unroll
  for (int c = 0; c < 3; ++c) {
    float mi = S[c] * invN;
    float mpi = S[4 + c] * invN;
    float mii = S[7 + c] * invN;
    float a = (mpi - mp * mi) / (mii - mi * mi + EPSG);
    float b = mp - a * mi;
    float rt = fminf(fmaxf(a * iv[c] + b, 0.f), 1.f);
    rt = fmaxf(rt, 0.1f);
    r[c] = fminf(fmaxf((iv[c] - avg) / rt + avg, 0.f), 1.f);
  }
  o0 = r[0]; o1 = r[1]; o2 = r[2];
}

__global__ void k_vfinal(const float* __restrict__ x, const float* __restrict__ planes,
                         const float* __restrict__ wsf, float* __restrict__ out) {
  const int xc = (blockIdx.x * 128 + threadIdx.x) * 2;  // even column
  const int y0 = blockIdx.y * 128;
  const float avg = wsf[1];
  float Sa[10], Sb[10];
#pragma unroll
  for (int q = 0; q < 10; ++q) { Sa[q] = 0.f; Sb[q] = 0.f; }
  const int lo = imax(y0 - RGUIDE, 0);
  const int hiy = imin(y0 + RGUIDE, HH - 1);
  for (int yy = lo; yy <= hiy; ++yy) {
    const float* row = planes + yy * WW + xc;
#pragma unroll
    for (int q = 0; q < 10; ++q) {
      float2 r = *(const float2*)(row + q * HWTOT);
      Sa[q] += r.x; Sb[q] += r.y;
    }
  }
  const float nx0 = (float)(imin(xc + RGUIDE, WW - 1) - imax(xc - RGUIDE, 0) + 1);
  const float nx1 = (float)(imin(xc + 1 + RGUIDE, WW - 1) - imax(xc + 1 - RGUIDE, 0) + 1);
  for (int y = y0; y < y0 + 128; ++y) {
    const float nyf = (float)(imin(y + RGUIDE, HH - 1) - imax(y - RGUIDE, 0) + 1);
    const float invN0 = 1.f / (nx0 * nyf);
    const float invN1 = 1.f / (nx1 * nyf);
    float2 i0 = *(const float2*)(x + 0 * HWTOT + y * WW + xc);
    float2 i1 = *(const float2*)(x + 1 * HWTOT + y * WW + xc);
    float2 i2 = *(const float2*)(x + 2 * HWTOT + y * WW + xc);
    float2 o0, o1, o2;
    dehaze_px(Sa, invN0, avg, i0.x, i1.x, i2.x, o0.x, o1.x, o2.x);
    dehaze_px(Sb, invN1, avg, i0.y, i1.y, i2.y, o0.y, o1.y, o2.y);
    *(float2*)(out + 0 * HWTOT + y * WW + xc) = o0;
    *(float2*)(out + 1 * HWTOT + y * WW + xc) = o1;
    *(float2*)(out + 2 * HWTOT + y * WW + xc) = o2;
    const int add = y + RGUIDE + 1;
    const int sub = y - RGUIDE;
    if (add < HH) {
      const float* row = planes + add * WW + xc;
#pragma unroll
      for (int q = 0; q < 10; ++q) {
        float2 r = *(const float2*)(row + q * HWTOT);
        Sa[q] += r.x; Sb[q] += r.y;
      }
    }
    if (sub >= 0) {
      const float* row = planes + sub * WW + xc;
#pragma unroll
      for (int q = 0; q < 10; ++q) {
        float2 r = *(const float2*)(row + q * HWTOT);
        Sa[q] -= r.x; Sb[q] -= r.y;
      }
    }
    if (((y & 7) == 0) && (add + 16 < HH)) {  // global_prefetch_b8, stream-ahead
      const float* row = planes + (add + 16) * WW + xc;
#pragma unroll
      for (int q = 0; q < 10; ++q) __builtin_prefetch(row + q * HWTOT, 0, 0);
    }
  }
}

// ---------------- host ----------------
extern "C" void kernel_launch(void* const* d_in, const int* in_sizes, int n_in,
                              void* d_out, int out_size, void* d_ws, size_t ws_size,
                              hipStream_t stream) {
  (void)in_sizes; (void)n_in; (void)out_size; (void)ws_size;
  const float* x = (const float*)d_in[0];
  float* out = (float*)d_out;
  float* wsf = (float*)d_ws;
  unsigned* hist = (unsigned*)(wsf + 256);   // 65536 bins
  float* dc = wsf + (1u << 18);              // 1 MiB offset
  float* planes = dc + HWTOT;                // 10 planes, 160 MiB
  float* tmp = planes;                       // dch_h reuses plane 0
  float* cmin = planes + HWTOT;              // channel-min reuses plane 1
  float* acc = wsf + 2;                      // cnt, s0, s1, s2

  {
    int n = 256 + 65536;  // scalars + histogram
    k_zero<<<(n + 255) / 256, 256, 0, stream>>>((unsigned*)wsf, n);
  }
  k_cmin<<<HWTOT / (4 * 256), 256, 0, stream>>>(x, cmin);
  k_hmin<<<HWTOT / 256, 256, 0, stream>>>(cmin, tmp);
  k_vmin<<<HWTOT / 256, 256, 0, stream>>>(tmp, dc);
  k_hist<<<HWTOT / (256 * 8), 256, 0, stream>>>(dc, hist);
  k_thresh<<<1, 256, 0, stream>>>(hist, wsf);
  k_masked<<<HWTOT / (256 * 16), 256, 0, stream>>>(x, dc, wsf, acc);
  k_avg<<<1, 1, 0, stream>>>(wsf);
  k_hbox<<<HH, 256, 0, stream>>>(x, dc, wsf, planes);
  k_vfinal<<<dim3(WW / 256, HH / 128), 128, 0, stream>>>(x, planes, wsf, out);
}